// ImprovedYOLOLoss_36936718746136
// MI455X (gfx1250) — compile-verified
//
#include <hip/hip_runtime.h>
#include <hip/hip_bf16.h>

// ---- problem constants ---------------------------------------------------
#define BATCH       128
#define HH          80
#define WW          80
#define HW          (HH * WW)          // 6400
#define NANCH       3
#define TOTAL_CELLS (BATCH * NANCH * HW)   // 2,457,600
#define NTGT        32

#define EPS_F       1e-7f
#define LAMBDA_COORD 5.0f
#define LAMBDA_OBJ   1.0f
#define LAMBDA_NOOBJ 0.5f

// partial-sum layout in workspace: 5 arrays of 1024 floats each
// q0 = noobj (dense sum + negative corrections), q1 = obj, q2 = xy, q3 = wh, q4 = count
#define DENSE_BLOCKS 896
#define TGT_BLOCKS   128
#define NPART        (DENSE_BLOCKS + TGT_BLOCKS)   // 1024

typedef __attribute__((ext_vector_type(2))) float v2f;
typedef __attribute__((ext_vector_type(8))) float v8f;

__device__ __forceinline__ float sigmoidf_(float x) {
    return 1.0f / (1.0f + __expf(-x));
}

__device__ __forceinline__ float noobj_term(float x) {
    float s  = sigmoidf_(x);
    float pc = fminf(fmaxf(s, EPS_F), 1.0f - EPS_F);
    return -__logf(1.0f - pc);
}

__device__ __forceinline__ float wave_sum32(float v) {
    #pragma unroll
    for (int off = 16; off > 0; off >>= 1)
        v += __shfl_down(v, off, 32);
    return v;
}

// ---- kernel 1: dense scan over all confidence planes ---------------------
// reads only channel (a*6+4) of each (b,a) pair: 384 planes * 6400 floats
__global__ __launch_bounds__(256)
void yolo_dense_conf(const float4* __restrict__ pred4, float* __restrict__ P) {
    const int total_g = BATCH * NANCH * (HW / 4);   // 614,400 float4 groups
    const int stride  = gridDim.x * blockDim.x;
    float acc = 0.0f;
    for (int g = blockIdx.x * blockDim.x + threadIdx.x; g < total_g; g += stride) {
        int plane = g / (HW / 4);
        int rem   = g - plane * (HW / 4);
        int b     = plane / NANCH;
        int a     = plane - b * NANCH;
        int base  = (b * 18 + a * 6 + 4) * (HW / 4) + rem;
        float4 x  = pred4[base];
        acc += noobj_term(x.x) + noobj_term(x.y) + noobj_term(x.z) + noobj_term(x.w);
    }
    // block reduction (8 waves)
    __shared__ float sm[8];
    int lane = threadIdx.x & 31;
    int wid  = threadIdx.x >> 5;
    float w  = wave_sum32(acc);
    if (lane == 0) sm[wid] = w;
    __syncthreads();
    if (threadIdx.x == 0) {
        float s = 0.0f;
        #pragma unroll
        for (int i = 0; i < 8; ++i) s += sm[i];
        int blk = blockIdx.x;
        P[0 * NPART + blk] = s;     // dense noobj partial
        P[1 * NPART + blk] = 0.0f;  // obj
        P[2 * NPART + blk] = 0.0f;  // xy
        P[3 * NPART + blk] = 0.0f;  // wh
        P[4 * NPART + blk] = 0.0f;  // count
    }
}

// ---- kernel 2: per-batch target processing (one wave per batch) ----------
__global__ __launch_bounds__(32)
void yolo_targets(const float* __restrict__ pred, const float* __restrict__ tg,
                  float* __restrict__ P) {
    int b = blockIdx.x;          // 0..127
    int t = threadIdx.x;         // 0..31 (one lane per target)
    const float* tr = tg + (size_t)(b * NTGT + t) * 6;
    float cx = tr[2], cy = tr[3], tw = tr[4], th = tr[5];
    float gx = cx * (float)WW;
    float gy = cy * (float)HH;
    float gxc = fminf(fmaxf(gx, 0.0f), (float)(WW - 1));
    float gyc = fminf(fmaxf(gy, 0.0f), (float)(HH - 1));
    int gi = (int)gxc;
    int gj = (int)gyc;
    int cell = gj * WW + gi;

    // in-wave duplicate resolution: "last write wins" for values,
    // first occurrence counts the unique cell.
    bool is_last = true, is_first = true;
    #pragma unroll
    for (int j = 0; j < 32; ++j) {
        int cj = __shfl(cell, j, 32);
        if (cj == cell) {
            if (j > t) is_last  = false;
            if (j < t) is_first = false;
        }
    }

    float xy = 0.0f, wh = 0.0f, obj = 0.0f, ncorr = 0.0f;
    float cnt = is_first ? 1.0f : 0.0f;
    if (is_last) {
        // anchor 0 only: channels 0..4 at (b, c, cell)
        const float* pb = pred + (size_t)b * 18 * HW + cell;
        float px  = pb[0 * HW];
        float py  = pb[1 * HW];
        float pw  = pb[2 * HW];
        float ph  = pb[3 * HW];
        float pcf = pb[4 * HW];
        float sx = sigmoidf_(px);
        float sy = sigmoidf_(py);
        float tx = gx - (float)gi;
        float ty = gy - (float)gj;
        xy = (sx - tx) * (sx - tx) + (sy - ty) * (sy - ty);
        wh = (pw - tw) * (pw - tw) + (ph - th) * (ph - th);
        float pc = fminf(fmaxf(sigmoidf_(pcf), EPS_F), 1.0f - EPS_F);
        obj   = -__logf(pc);
        ncorr =  __logf(1.0f - pc);   // removes this cell from dense noobj sum
    }

    float ncorr_s = wave_sum32(ncorr);
    float obj_s   = wave_sum32(obj);
    float xy_s    = wave_sum32(xy);
    float wh_s    = wave_sum32(wh);
    float cnt_s   = wave_sum32(cnt);
    if (t == 0) {
        int idx = DENSE_BLOCKS + b;
        P[0 * NPART + idx] = ncorr_s;
        P[1 * NPART + idx] = obj_s;
        P[2 * NPART + idx] = xy_s;
        P[3 * NPART + idx] = wh_s;
        P[4 * NPART + idx] = cnt_s;
    }
}

// ---- kernel 3: deterministic final reduction via WMMA --------------------
// V_WMMA_F32_16X16X4_F32 with B == ones(4x16):
//   D[m][n] = sum_k A[m][k] + C[m][n]  -> folds 64 partials per instruction,
//   every column of D identical, so a single shfl_xor(16) finishes the sum.
__global__ __launch_bounds__(32)
void yolo_final(const float* __restrict__ P, float* __restrict__ out) {
    int lane = threadIdx.x;      // exactly one wave32, EXEC all ones
    v2f onesB;
    onesB[0] = 1.0f; onesB[1] = 1.0f;

    float sums[5];
    #pragma unroll
    for (int q = 0; q < 5; ++q) {
        const float* Pq = P + q * NPART;
        v8f acc = {};
        #pragma unroll
        for (int c = 0; c < NPART / 64; ++c) {    // 16 chunks of 64 partials
            v2f a;
            a[0] = Pq[c * 64 + 2 * lane];
            a[1] = Pq[c * 64 + 2 * lane + 1];
            acc = __builtin_amdgcn_wmma_f32_16x16x4_f32(
                /*neg_a=*/false, a, /*neg_b=*/false, onesB,
                /*c_mod=*/(short)0, acc, /*reuse_a=*/false, /*reuse_b=*/false);
        }
        float v = acc[0] + acc[1] + acc[2] + acc[3] +
                  acc[4] + acc[5] + acc[6] + acc[7];
        v += __shfl_xor(v, 16, 32);   // lanes 0-15 hold M=0..7, 16-31 hold M=8..15
        sums[q] = v;
    }

    if (lane == 0) {
        float noobj_sum = sums[0];
        float obj_sum   = sums[1];
        float xy_sum    = sums[2];
        float wh_sum    = sums[3];
        float cnt       = sums[4];
        float num_noobj = (float)TOTAL_CELLS - cnt;
        float safe_obj   = fmaxf(cnt, 1.0f);
        float safe_noobj = fmaxf(num_noobj, 1.0f);
        float coord_l = (xy_sum + wh_sum) / safe_obj;
        float obj_l   = obj_sum / safe_obj;
        float noobj_l = noobj_sum / safe_noobj;
        float total = LAMBDA_COORD * (cnt > 0.0f ? coord_l : 0.0f)
                    + LAMBDA_OBJ   * (cnt > 0.0f ? obj_l   : 0.0f)
                    + LAMBDA_NOOBJ * (num_noobj > 0.0f ? noobj_l : 0.0f);
        out[0] = total;
    }
}

// ---- host-side launch ----------------------------------------------------
extern "C" void kernel_launch(void* const* d_in, const int* in_sizes, int n_in,
                              void* d_out, int out_size, void* d_ws, size_t ws_size,
                              hipStream_t stream) {
    (void)in_sizes; (void)n_in; (void)out_size; (void)ws_size;
    const float* pred    = (const float*)d_in[0];
    const float* targets = (const float*)d_in[1];
    float*       out     = (float*)d_out;
    float*       P       = (float*)d_ws;   // 5 * 1024 floats = 20 KB

    yolo_dense_conf<<<DENSE_BLOCKS, 256, 0, stream>>>((const float4*)pred, P);
    yolo_targets<<<TGT_BLOCKS, 32, 0, stream>>>(pred, targets, P);
    yolo_final<<<1, 32, 0, stream>>>(P, out);
}